// ReadinCrossAttention_5574867550883
// MI455X (gfx1250) — compile-verified
//
#include <hip/hip_runtime.h>
#include <math.h>

typedef __attribute__((ext_vector_type(2))) float v2f;
typedef __attribute__((ext_vector_type(8))) float v8f;

#define R_DIM 128
#define H_DIM 128
#define C_DIM 128
#define OUT_DIM 256
#define TA 512          // T*A
#define NROWS 4096      // B*T*A

// workspace layout (float offsets)
#define OFF_POSK  0        // [128][128]
#define OFF_POSV  16384    // [128][128]
#define OFF_SESSK 32768    // [8][128]
#define OFF_SESSV 33792    // [8][128]
#define OFF_KQ    34816    // [128]
#define OFF_KBQ   34944    // [1]
#define OFF_W     34945    // [8][128]

__device__ __forceinline__ float gelu_exact(float x) {
  // jax.nn.gelu(approximate=False): 0.5*x*(1+erf(x/sqrt(2)))
  return 0.5f * x * (1.0f + erff(x * 0.70710678118654752f));
}

// posK[c][r] = sum_k pos[c][k]*kW1[k][r]; posV likewise with vW1
__global__ void pos_proj_kernel(const float* __restrict__ pos, const float* __restrict__ kW1,
                                const float* __restrict__ vW1, float* __restrict__ ws) {
  int c = blockIdx.x, r = threadIdx.x;
  float a1 = 0.f, a2 = 0.f;
  for (int k = 0; k < R_DIM; ++k) {
    float p = pos[c * R_DIM + k];
    a1 = fmaf(p, kW1[k * R_DIM + r], a1);
    a2 = fmaf(p, vW1[k * R_DIM + r], a2);
  }
  ws[OFF_POSK + c * R_DIM + r] = a1;
  ws[OFF_POSV + c * R_DIM + r] = a2;
}

// sessK[b][r] = sum_k session[b][k]*kW1[R+k][r]; sessV likewise with vW1
__global__ void sess_proj_kernel(const float* __restrict__ session, const float* __restrict__ kW1,
                                 const float* __restrict__ vW1, float* __restrict__ ws) {
  int b = blockIdx.x, r = threadIdx.x;
  float a1 = 0.f, a2 = 0.f;
  for (int k = 0; k < H_DIM; ++k) {
    float s = session[b * H_DIM + k];
    a1 = fmaf(s, kW1[(R_DIM + k) * R_DIM + r], a1);
    a2 = fmaf(s, vW1[(R_DIM + k) * R_DIM + r], a2);
  }
  ws[OFF_SESSK + b * R_DIM + r] = a1;
  ws[OFF_SESSV + b * R_DIM + r] = a2;
}

// kq[r] = kW2[r,:] . query ; kbq = kb2 . query
__global__ void kq_kernel(const float* __restrict__ kW2, const float* __restrict__ kb2,
                          const float* __restrict__ query, float* __restrict__ ws) {
  int r = threadIdx.x;
  float a = 0.f;
  for (int j = 0; j < R_DIM; ++j) a = fmaf(kW2[r * R_DIM + j], query[j], a);
  ws[OFF_KQ + r] = a;
  if (r == 0) {
    float s = 0.f;
    for (int j = 0; j < R_DIM; ++j) s = fmaf(kb2[j], query[j], s);
    ws[OFF_KBQ] = s;
  }
}

// scores[b][c] = (sum_r gelu(posK+sessK+kb1)*kq[r] + kbq)/sqrt(R); softmax over c -> w
__global__ void softmax_kernel(const float* __restrict__ kb1, float* __restrict__ ws) {
  int b = blockIdx.x, c = threadIdx.x;
  const float* posK  = ws + OFF_POSK + c * R_DIM;
  const float* sessK = ws + OFF_SESSK + b * R_DIM;
  const float* kq    = ws + OFF_KQ;
  float acc = 0.f;
  for (int r = 0; r < R_DIM; ++r) {
    float x = posK[r] + sessK[r] + kb1[r];
    acc = fmaf(gelu_exact(x), kq[r], acc);
  }
  float score = (acc + ws[OFF_KBQ]) * 0.08838834764831845f; // 1/sqrt(128)
  __shared__ float red[C_DIM];
  red[c] = score; __syncthreads();
  for (int off = 64; off > 0; off >>= 1) {
    if (c < off) red[c] = fmaxf(red[c], red[c + off]);
    __syncthreads();
  }
  float mx = red[0]; __syncthreads();
  float e = expf(score - mx);
  red[c] = e; __syncthreads();
  for (int off = 64; off > 0; off >>= 1) {
    if (c < off) red[c] += red[c + off];
    __syncthreads();
  }
  ws[OFF_W + b * C_DIM + c] = e / red[0];
}

// Main kernel: 16 output rows per block.
//  phase 1: g[i][r] = sum_c w[c]*gelu(base[c][r] + s[i][c]*u[r])   (erf-bound)
//  phase 2: out[16x256] = g[16x128] @ vW2[128x256] + vb2  via v_wmma_f32_16x16x4_f32
__global__ void __launch_bounds__(256)
values_attn_kernel(const float* __restrict__ state_in, const float* __restrict__ vb1,
                   const float* __restrict__ vW1, const float* __restrict__ vW2,
                   const float* __restrict__ vb2, const float* __restrict__ ws,
                   float* __restrict__ out) {
  __shared__ float baseS[32 * R_DIM];  // 16 KB: chunk of 32 c-values
  __shared__ float sS[16 * C_DIM];     // 8 KB: state tile
  __shared__ float gS[16 * R_DIM];     // 8 KB: weighted-gelu result
  __shared__ float wS[C_DIM];          // softmax weights

  int tid  = threadIdx.x;
  int row0 = blockIdx.x * 16;          // 16 rows, all within one b (TA=512 divisible by 16)
  int b    = row0 / TA;

  const float* posV  = ws + OFF_POSV;
  const float* sessV = ws + OFF_SESSV + b * R_DIM;
  const float* wptr  = ws + OFF_W + b * C_DIM;
  const float* uvec  = vW1 + (R_DIM + H_DIM) * R_DIM; // vW1 row 256 (state_in column)

  if (tid < C_DIM) wS[tid] = wptr[tid];
  for (int idx = tid; idx < 16 * C_DIM; idx += 256)
    sS[idx] = state_in[row0 * C_DIM + idx];

  int r  = tid & 127;   // this thread's r
  int i0 = tid >> 7;    // 0 or 1; handles rows i0, i0+2, ..., i0+14
  float uu = uvec[r];

  float accv[8];
  #pragma unroll
  for (int p = 0; p < 8; ++p) accv[p] = 0.f;

  for (int chunk = 0; chunk < 4; ++chunk) {
    __syncthreads(); // previous chunk consumed (covers initial sS/wS fill on chunk 0)
    int cbase = chunk * 32;
    for (int idx = tid; idx < 32 * R_DIM; idx += 256) {
      int cc = idx >> 7, rr = idx & 127;
      baseS[idx] = posV[(cbase + cc) * R_DIM + rr] + sessV[rr] + vb1[rr];
    }
    __syncthreads();
    for (int cc = 0; cc < 32; ++cc) {
      float bval = baseS[cc * R_DIM + r];   // conflict-free: bank = r%64
      float wgt  = wS[cbase + cc];          // broadcast
      #pragma unroll
      for (int p = 0; p < 8; ++p) {
        int i = i0 + 2 * p;
        float x = fmaf(sS[i * C_DIM + cbase + cc], uu, bval);
        accv[p] = fmaf(wgt, gelu_exact(x), accv[p]);
      }
    }
  }

  #pragma unroll
  for (int p = 0; p < 8; ++p) gS[(i0 + 2 * p) * R_DIM + r] = accv[p];
  __syncthreads();

  // ---- WMMA phase: 8 waves; wave handles N-tiles {2*wave, 2*wave+1} ----
  int lane   = tid & 31;
  int waveId = tid >> 5;
  int m      = lane & 15;            // A row
  int khalf  = (lane >> 4) << 1;     // 0 (lanes 0-15) or 2 (lanes 16-31)
  int nloc   = lane & 15;            // B/D column within tile
  int n0     = waveId * 32 + nloc;   // global output column (tile 0)

  v8f acc0 = {0.f,0.f,0.f,0.f,0.f,0.f,0.f,0.f};
  v8f acc1 = {0.f,0.f,0.f,0.f,0.f,0.f,0.f,0.f};
  const float* gRow = &gS[m * R_DIM + khalf];

  for (int s = 0; s < 32; ++s) {
    int k0 = s * 4;
    v2f a;                           // A[m][k0+khalf], A[m][k0+khalf+1]
    a.x = gRow[k0];
    a.y = gRow[k0 + 1];
    int kb = k0 + khalf;
    v2f b0, b1;                      // B[k][n] symmetric layout to A
    b0.x = vW2[kb * OUT_DIM + n0];
    b0.y = vW2[(kb + 1) * OUT_DIM + n0];
    b1.x = vW2[kb * OUT_DIM + n0 + 16];
    b1.y = vW2[(kb + 1) * OUT_DIM + n0 + 16];
    acc0 = __builtin_amdgcn_wmma_f32_16x16x4_f32(false, a, false, b0, (short)0, acc0, false, false);
    acc1 = __builtin_amdgcn_wmma_f32_16x16x4_f32(false, a, false, b1, (short)0, acc1, false, false);
  }

  // D layout: VGPR v -> row = v + 8*(lane/16), col = lane%16 within tile
  int mr = (lane >> 4) * 8;
  float bias0 = vb2[n0], bias1 = vb2[n0 + 16];
  #pragma unroll
  for (int v = 0; v < 8; ++v) {
    int row = row0 + mr + v;
    out[row * OUT_DIM + n0]      = acc0[v] + bias0;
    out[row * OUT_DIM + n0 + 16] = acc1[v] + bias1;
  }
}

extern "C" void kernel_launch(void* const* d_in, const int* in_sizes, int n_in,
                              void* d_out, int out_size, void* d_ws, size_t ws_size,
                              hipStream_t stream) {
  const float* state_in = (const float*)d_in[0];
  const float* session  = (const float*)d_in[1];
  // d_in[2] subject, d_in[3] array: unused by the reference
  const float* query    = (const float*)d_in[4];
  const float* pos      = (const float*)d_in[5];
  const float* kW1      = (const float*)d_in[6];
  const float* kb1      = (const float*)d_in[7];
  const float* kW2      = (const float*)d_in[8];
  const float* kb2      = (const float*)d_in[9];
  const float* vW1      = (const float*)d_in[10];
  const float* vb1      = (const float*)d_in[11];
  const float* vW2      = (const float*)d_in[12];
  const float* vb2      = (const float*)d_in[13];
  float* out = (float*)d_out;
  float* ws  = (float*)d_ws;

  pos_proj_kernel<<<128, 128, 0, stream>>>(pos, kW1, vW1, ws);
  sess_proj_kernel<<<8, 128, 0, stream>>>(session, kW1, vW1, ws);
  kq_kernel<<<1, 128, 0, stream>>>(kW2, kb2, query, ws);
  softmax_kernel<<<8, 128, 0, stream>>>(kb1, ws);
  values_attn_kernel<<<NROWS / 16, 256, 0, stream>>>(state_in, vb1, vW1, vW2, vb2, ws, out);
}